// GINLayer_14594298871931
// MI455X (gfx1250) — compile-verified
//
#include <hip/hip_runtime.h>
#include <hip/hip_bf16.h>

typedef __attribute__((ext_vector_type(2))) float v2f;
typedef __attribute__((ext_vector_type(8))) float v8f;

#define IN_FEATS 256
#define HIDDEN   512
#define OUT_FEATS 256
#define EPS 0.0f

// ---------------------------------------------------------------------------
// Kernel 1: agg[i] = (1+eps) * x[i]   (float4 vectorized, re-inits scratch)
// ---------------------------------------------------------------------------
__global__ void k_init(const float4* __restrict__ x4, float4* __restrict__ agg4,
                       int n4) {
  int i = blockIdx.x * blockDim.x + threadIdx.x;
  if (i < n4) {
    float4 v = x4[i];
    const float s = 1.0f + EPS;
    v.x *= s; v.y *= s; v.z *= s; v.w *= s;
    agg4[i] = v;
  }
}

// ---------------------------------------------------------------------------
// Kernel 2: edge scatter-add. One wave32 per edge; lane l handles features
// l, l+32, ..., l+224 so each atomic wave-op covers one contiguous 128B line.
// Native fp32 HW atomic forced via inline asm (device scope: rows are shared
// across workgroups). x / agg are L2-resident (51.2 MB each << 192 MB L2).
// ---------------------------------------------------------------------------
__device__ __forceinline__ void hw_atomic_add_f32(float* p, float v) {
  asm volatile("global_atomic_add_f32 %0, %1, off scope:SCOPE_DEV"
               :: "v"(p), "v"(v) : "memory");
}

__global__ void k_scatter(const float* __restrict__ x,
                          const int* __restrict__ erow,
                          const int* __restrict__ ecol,
                          float* __restrict__ agg, int n_edges) {
  int gid  = blockIdx.x * blockDim.x + threadIdx.x;
  int edge = gid >> 5;
  int lane = gid & 31;
  if (edge < n_edges) {
    int r = erow[edge];
    int c = ecol[edge];
    const float* __restrict__ src = x   + (size_t)c * IN_FEATS;
    float*                    dst = agg + (size_t)r * IN_FEATS;
#pragma unroll
    for (int k = 0; k < IN_FEATS / 32; ++k) {
      int f = lane + 32 * k;
      hw_atomic_add_f32(&dst[f], src[f]);
    }
  }
}

// ---------------------------------------------------------------------------
// Kernel 3: fused MLP per 16-row tile, fp32 WMMA (V_WMMA_F32_16X16X4_F32).
//   sA (16x256 LDS, async-staged)  --GEMM1-->  sH (16x512 LDS)  --GEMM2--> out
// 256 threads = 8 waves; GEMM1: 4 N-tiles/wave; GEMM2: 2 N-tiles/wave.
// A-frag (16x4 f32): lane L -> M = L&15, K pair = (L>=16 ? {2,3} : {0,1}).
// C-frag: VGPR v, lane L -> M = v + 8*(L>>4), N = L&15  => bias pre-load.
// ---------------------------------------------------------------------------
__global__ void __launch_bounds__(256)
k_mlp(const float* __restrict__ agg,
      const float* __restrict__ W1, const float* __restrict__ b1,
      const float* __restrict__ W2, const float* __restrict__ b2,
      float* __restrict__ out, int n_nodes) {
  __shared__ float sA[16 * IN_FEATS];   // 16 KB
  __shared__ float sH[16 * HIDDEN];     // 32 KB

  const int tid  = threadIdx.x;
  const int wave = tid >> 5;
  const int lane = tid & 31;
  const int half = lane >> 4;      // 0: lanes 0-15, 1: lanes 16-31
  const int col  = lane & 15;      // M for A-frag, N for B/C-frag
  const int row0 = blockIdx.x * 16;

  // ---- stage agg tile into LDS via async global->LDS B128 (ASYNCcnt) ----
  {
    const float4* g = (const float4*)agg;
    float4*       s = (float4*)sA;
#pragma unroll
    for (int i = 0; i < 4; ++i) {
      int e  = tid + i * 256;             // float4 index within 16x64 tile
      int r  = e >> 6;                    // row in tile
      int c4 = e & 63;                    // float4 col
      int gr = row0 + r;
      if (gr >= n_nodes) gr = n_nodes - 1;
      const float4* gp = &g[(size_t)gr * (IN_FEATS / 4) + c4];
      unsigned lds_off = (unsigned)(size_t)(&s[e]);  // LDS byte offset
      asm volatile("global_load_async_to_lds_b128 %0, %1, off"
                   :: "v"(lds_off), "v"(gp) : "memory");
    }
    asm volatile("s_wait_asynccnt 0" ::: "memory");
  }
  __syncthreads();

  // ---- GEMM1: sH[16x512] = sA[16x256] @ W1 + b1 ----
#pragma unroll
  for (int t = 0; t < 4; ++t) {
    const int n0 = (wave * 4 + t) * 16;
    v8f acc;
    const float bias = b1[n0 + col];
#pragma unroll
    for (int v = 0; v < 8; ++v) acc[v] = bias;

    for (int k0 = 0; k0 < IN_FEATS; k0 += 4) {
      v2f a = *(const v2f*)&sA[col * IN_FEATS + k0 + 2 * half];
      v2f b;
      b.x = W1[(size_t)(k0 + 2 * half)     * HIDDEN + n0 + col];
      b.y = W1[(size_t)(k0 + 2 * half + 1) * HIDDEN + n0 + col];
      acc = __builtin_amdgcn_wmma_f32_16x16x4_f32(
          /*neg_a=*/false, a, /*neg_b=*/false, b,
          /*c_mod=*/(short)0, acc, /*reuse_a=*/false, /*reuse_b=*/false);
    }
#pragma unroll
    for (int v = 0; v < 8; ++v)
      sH[(v + 8 * half) * HIDDEN + n0 + col] = acc[v];
  }
  __syncthreads();

  // ---- GEMM2: out[16x256] = sH[16x512] @ W2 + b2 ----
#pragma unroll
  for (int t = 0; t < 2; ++t) {
    const int n0 = (wave * 2 + t) * 16;
    v8f acc;
    const float bias = b2[n0 + col];
#pragma unroll
    for (int v = 0; v < 8; ++v) acc[v] = bias;

    for (int k0 = 0; k0 < HIDDEN; k0 += 4) {
      v2f a = *(const v2f*)&sH[col * HIDDEN + k0 + 2 * half];
      v2f b;
      b.x = W2[(size_t)(k0 + 2 * half)     * OUT_FEATS + n0 + col];
      b.y = W2[(size_t)(k0 + 2 * half + 1) * OUT_FEATS + n0 + col];
      acc = __builtin_amdgcn_wmma_f32_16x16x4_f32(
          false, a, false, b, (short)0, acc, false, false);
    }
#pragma unroll
    for (int v = 0; v < 8; ++v) {
      int gr = row0 + v + 8 * half;
      if (gr >= n_nodes) gr = n_nodes - 1;
      out[(size_t)gr * OUT_FEATS + n0 + col] = acc[v];
    }
  }
}

// ---------------------------------------------------------------------------
// Launch
// ---------------------------------------------------------------------------
extern "C" void kernel_launch(void* const* d_in, const int* in_sizes, int n_in,
                              void* d_out, int out_size, void* d_ws, size_t ws_size,
                              hipStream_t stream) {
  const float* x    = (const float*)d_in[0];
  const int*   erow = (const int*)d_in[1];
  const int*   ecol = (const int*)d_in[2];
  const float* W1   = (const float*)d_in[3];
  const float* b1   = (const float*)d_in[4];
  const float* W2   = (const float*)d_in[5];
  const float* b2   = (const float*)d_in[6];
  float*       out  = (float*)d_out;
  float*       agg  = (float*)d_ws;          // n_nodes * 256 floats (51.2 MB)

  const int n_nodes = in_sizes[0] / IN_FEATS;
  const int n_edges = in_sizes[1];

  // 1) agg = (1+eps)*x
  {
    int n4 = n_nodes * (IN_FEATS / 4);
    int blocks = (n4 + 255) / 256;
    k_init<<<blocks, 256, 0, stream>>>((const float4*)x, (float4*)agg, n4);
  }

  // 2) scatter-add edges (one wave per edge, native f32 atomics)
  {
    long long threads = (long long)n_edges * 32;
    int blocks = (int)((threads + 255) / 256);
    k_scatter<<<blocks, 256, 0, stream>>>(x, erow, ecol, agg, n_edges);
  }

  // 3) fused 2-layer MLP via fp32 WMMA
  {
    int blocks = (n_nodes + 15) / 16;
    k_mlp<<<blocks, 256, 0, stream>>>(agg, W1, b1, W2, b2, out, n_nodes);
  }
}